// OneHotEmbedding_87522843561070
// MI455X (gfx1250) — compile-verified
//
#include <hip/hip_runtime.h>

// Embedding gather: out[i, 0:128] = emb[idx[i], 0:128]   (float32)
// One wave32 per row: 128 floats = 512 B = 32 lanes x float4 (global_load_b128).
// Two rows per wave with a global_prefetch_b8 of row1 issued before row0's
// load/store, to overlap latency. Output stored with non-temporal hint
// (write-once stream, never re-read by this kernel).

typedef __attribute__((ext_vector_type(4))) float v4f;

__global__ __launch_bounds__(256) void OneHotEmbedding_gather_kernel(
    const int* __restrict__ idx,
    const float* __restrict__ emb,
    float* __restrict__ out,
    int n_rows) {
  const int waves_per_block = blockDim.x >> 5;            // 8 for 256 threads
  const int wave = blockIdx.x * waves_per_block + (threadIdx.x >> 5);
  const int lane = threadIdx.x & 31;
  const int total_waves = gridDim.x * waves_per_block;

  const int r0 = wave;                 // first row for this wave
  const int r1 = wave + total_waves;   // second row for this wave
  if (r0 >= n_rows) return;

  // Row r0: wave-uniform index load (single 4B transaction), then float4 gather.
  const int i0 = idx[r0];
  const v4f* __restrict__ src0 = (const v4f*)emb + (size_t)i0 * 32 + lane;
  v4f* __restrict__ dst0 = (v4f*)out + (size_t)r0 * 32 + lane;

  if (r1 < n_rows) {
    const int i1 = idx[r1];
    const v4f* __restrict__ src1 = (const v4f*)emb + (size_t)i1 * 32 + lane;
    v4f* __restrict__ dst1 = (v4f*)out + (size_t)r1 * 32 + lane;

    // Kick row1's cacheline toward L2/L0 while we move row0.
    __builtin_prefetch(src1, 0, 1);   // -> global_prefetch_b8 on gfx1250

    v4f d0 = *src0;                    // global_load_b128 (512 B / wave, coalesced)
    __builtin_nontemporal_store(d0, dst0);  // global_store_b128 th:NT

    v4f d1 = *src1;
    __builtin_nontemporal_store(d1, dst1);
  } else {
    v4f d0 = *src0;
    __builtin_nontemporal_store(d0, dst0);
  }
}

extern "C" void kernel_launch(void* const* d_in, const int* in_sizes, int n_in,
                              void* d_out, int out_size, void* d_ws, size_t ws_size,
                              hipStream_t stream) {
  const int*   idx = (const int*)d_in[0];    // [N] int32 token ids
  const float* emb = (const float*)d_in[1];  // [32000, 128] float32
  float*       out = (float*)d_out;          // [N, 128] float32

  const int n = in_sizes[0];                 // N = 16384

  const int threads = 256;                   // 8 wave32 per block
  const int waves_per_block = threads / 32;
  const int rows_per_wave = 2;
  const int total_waves = (n + rows_per_wave - 1) / rows_per_wave;
  const int blocks = (total_waves + waves_per_block - 1) / waves_per_block;

  OneHotEmbedding_gather_kernel<<<blocks, threads, 0, stream>>>(idx, emb, out, n);
}